// Polarisation_71356586656299
// MI455X (gfx1250) — compile-verified
//
#include <hip/hip_runtime.h>
#include <cstdint>
#include <cstddef>

// Toggle for the CDNA5 async global->LDS path (ASYNCcnt-tracked DMA-style copy).
#ifndef USE_ASYNC_LDS
#define USE_ASYNC_LDS 1
#endif

typedef __attribute__((ext_vector_type(4))) float f4;

#define BOHR_F      0.52917721067f
#define INV_BOHR_F  (1.0f / BOHR_F)
#define A_MUTUAL_F  0.39f
#define LOG2E_F     1.4426950408889634f

// One block = 256 threads = 8 wave32 = 256 edges.
// Memory-bound streaming kernel: ~60 B/edge, 384 MB total -> ~16.5 us at 23.3 TB/s.
__global__ __launch_bounds__(256) void thole_tmatrix_kernel(
    const int*   __restrict__ edge_src,
    const int*   __restrict__ edge_dst,
    const float* __restrict__ distances,
    const float* __restrict__ vec,
    const float* __restrict__ pol,
    float*       __restrict__ out,
    int nE)
{
    __shared__ __align__(16) float smem[256 * 9];

    const int t    = threadIdx.x;
    const int e0   = blockIdx.x * 256;
    const int e    = e0 + t;
    const bool full = (e0 + 256) <= nE;
    const int nloc  = full ? 256 : (nE - e0);

    // ---------------- Stage vec tile (768 floats, 16B aligned) into LDS ----------------
    const float* vg = vec + (size_t)e0 * 3;
#if USE_ASYNC_LDS
    if (full) {
        if (t < 192) {
            uint64_t ga = (uint64_t)(uintptr_t)(vg + t * 4);      // 16B per lane
            uint32_t la = (uint32_t)(uintptr_t)(&smem[t * 4]);    // LDS byte offset (low 32 bits)
            asm volatile("global_load_async_to_lds_b128 %0, %1, off"
                         :: "v"(la), "v"(ga) : "memory");
        }
        // per-wave wait on ASYNCcnt, then cross-wave barrier
        asm volatile("s_wait_asynccnt 0" ::: "memory");
    } else {
        const int nf = nloc * 3;
        for (int i = t; i < nf; i += 256) smem[i] = vg[i];
    }
#else
    {
        const int nf = nloc * 3;
        for (int i = t; i < nf; i += 256) smem[i] = vg[i];
    }
#endif
    __syncthreads();

    float vx = 0.f, vy = 0.f, vz = 0.f;
    if (t < nloc) {
        vx = smem[t * 3 + 0];
        vy = smem[t * 3 + 1];
        vz = smem[t * 3 + 2];
    }
    __syncthreads();   // smem is reused for the output tile below

    // ---------------- Per-edge Thole tensor ----------------
    if (t < nloc) {
        float d  = distances[e];
        int   si = edge_src[e];
        int   di = edge_dst[e];
        float pp = pol[si] * pol[di];              // pol gathers: 400 KB table, L2-resident

        // u = r_au / alpha_au^(1/6) == d / (p_i*p_j)^(1/6)  (BOHR factors cancel)
        // x^(-1/6) = exp2(-log2(x)/6) via v_log_f32 / v_exp_f32
        float u   = d * __builtin_amdgcn_exp2f(__builtin_amdgcn_logf(pp) * (-1.0f / 6.0f));
        float au3 = A_MUTUAL_F * u * u * u;
        float ex  = __builtin_amdgcn_exp2f(-au3 * LOG2E_F);   // exp(-au3)
        float l3  = 1.0f - ex;
        float l5  = 1.0f - (1.0f + au3) * ex;

        float rij = d * INV_BOHR_F;
        float ir  = 1.0f / rij;
        float ir2 = ir * ir;
        float ir3 = ir2 * ir;
        float ir5 = ir3 * ir2;
        float c5  = 3.0f * l5 * ir5;
        float c3  = l3 * ir3;

        float ax = vx * INV_BOHR_F;
        float ay = vy * INV_BOHR_F;
        float az = vz * INV_BOHR_F;

        float* o = &smem[t * 9];                   // stride 9: gcd(9,64)=1 -> conflict-free
        o[0] = c5 * ax * ax - c3;
        o[1] = c5 * ax * ay;
        o[2] = c5 * ax * az;
        o[3] = c5 * ay * ax;
        o[4] = c5 * ay * ay - c3;
        o[5] = c5 * ay * az;
        o[6] = c5 * az * ax;
        o[7] = c5 * az * ay;
        o[8] = c5 * az * az - c3;
    }
    __syncthreads();

    // ---------------- Coalesced non-temporal b128 output (2304 floats / block) ----------------
    float* outp = out + (size_t)e0 * 9;            // e0*9 divisible by 4 -> 16B aligned
    if (full) {
        f4*       o4 = (f4*)outp;
        const f4* s4 = (const f4*)smem;
        __builtin_nontemporal_store(s4[t],        &o4[t]);
        __builtin_nontemporal_store(s4[t + 256],  &o4[t + 256]);
        if (t < 64)
            __builtin_nontemporal_store(s4[t + 512], &o4[t + 512]);
    } else {
        const int nf = nloc * 9;
        for (int i = t; i < nf; i += 256)
            __builtin_nontemporal_store(smem[i], &outp[i]);
    }
}

extern "C" void kernel_launch(void* const* d_in, const int* in_sizes, int n_in,
                              void* d_out, int out_size, void* d_ws, size_t ws_size,
                              hipStream_t stream) {
    // setup_inputs order: species(0), edge_src(1), edge_dst(2), distances(3), vec(4), polarisability(5)
    const int*   edge_src  = (const int*)  d_in[1];
    const int*   edge_dst  = (const int*)  d_in[2];
    const float* distances = (const float*)d_in[3];
    const float* vec       = (const float*)d_in[4];
    const float* pol       = (const float*)d_in[5];
    float*       out       = (float*)d_out;

    const int nE = in_sizes[3];
    const int blocks = (nE + 255) / 256;
    thole_tmatrix_kernel<<<blocks, 256, 0, stream>>>(
        edge_src, edge_dst, distances, vec, pol, out, nE);
}